// JointOrchestrator_26216480375216
// MI455X (gfx1250) — compile-verified
//
#include <hip/hip_runtime.h>
#include <math.h>

// Problem constants (match reference setup_inputs)
#define T_TOK   131072
#define H_DIM   768
#define G_DIM   512
#define N_SENT  4096
#define P_TOK   8
#define OUT_W   (H_DIM + G_DIM)   // 1280
#define L2_EPS  1e-12f

typedef float v2f __attribute__((ext_vector_type(2)));
typedef float v4f __attribute__((ext_vector_type(4)));   // native vector: OK for NT builtins
typedef float v8f __attribute__((ext_vector_type(8)));

__device__ __forceinline__ float wave_reduce_sum(float v) {
    // full butterfly: every lane ends with the total (wave32)
    #pragma unroll
    for (int m = 16; m > 0; m >>= 1) v += __shfl_xor(v, m, 32);
    return v;
}

// ---------------------------------------------------------------------------
// Kernel A: one workgroup, 1024 threads = 32 waves.
//  1) txt = mean(prompt_tokens, axis=0)                       (threads 0..767)
//  2) gnn = W @ txt + b  via V_WMMA_F32_16X16X4_F32           (wave w -> rows 16w..16w+15)
//  3) l2-normalize gnn; l2-normalize prompt rows; write the 8 prompt output rows
// ---------------------------------------------------------------------------
__global__ void __launch_bounds__(1024)
prompt_wmma_kernel(const float* __restrict__ prompt,   // [P, H]
                   const float* __restrict__ W,        // [G, H]
                   const float* __restrict__ b,        // [G]
                   float* __restrict__ out)            // [T+P, OUT_W]
{
    __shared__ float s_txt[H_DIM];
    __shared__ float s_gnn[G_DIM];
    __shared__ float s_ss;

    const int tid  = threadIdx.x;
    const int wid  = tid >> 5;
    const int lane = tid & 31;

    // ---- 1) column mean over the 8 prompt rows ----
    if (tid < H_DIM) {
        float s = 0.f;
        #pragma unroll
        for (int p = 0; p < P_TOK; ++p) s += prompt[p * H_DIM + tid];
        s_txt[tid] = s * (1.0f / (float)P_TOK);
    }
    if (tid == 0) s_ss = 0.f;
    __syncthreads();

    // ---- 2) WMMA matvec: wave w computes gnn[16w .. 16w+15] ----
    // A layout (16x4 f32): lanes 0-15 -> M=lane, VGPR{0,1}=K{0,1};
    //                      lanes 16-31 -> M=lane-16, VGPR{0,1}=K{2,3}.
    // B replicated across all N columns -> every D column equals W_tile @ txt.
    {
        const int m     = (wid << 4) + (lane & 15);     // row of W this lane feeds
        const int khalf = (lane >> 4);                   // 0 or 1
        const float* wrow = W + (size_t)m * H_DIM;

        v8f acc = {};
        for (int k = 0; k < H_DIM; k += 4) {
            const int kk = k + 2 * khalf;
            v2f a, bb;
            a.x  = wrow[kk];     a.y  = wrow[kk + 1];
            bb.x = s_txt[kk];    bb.y = s_txt[kk + 1];
            acc = __builtin_amdgcn_wmma_f32_16x16x4_f32(
                    false, a, false, bb, (short)0, acc, false, false);
        }
        // D layout: VGPR r -> (M = 16w + r, lanes 0-15) / (M = 16w + 8 + r, lanes 16-31)
        if (lane == 0) {
            #pragma unroll
            for (int r = 0; r < 8; ++r) {
                const int g = (wid << 4) + r;
                s_gnn[g] = acc[r] + b[g];
            }
        }
        if (lane == 16) {
            #pragma unroll
            for (int r = 0; r < 8; ++r) {
                const int g = (wid << 4) + 8 + r;
                s_gnn[g] = acc[r] + b[g];
            }
        }
    }
    __syncthreads();

    // ---- 3a) ||gnn|| via per-wave partials + one LDS atomic per wave ----
    if (tid < G_DIM) {
        const float v = s_gnn[tid];
        const float part = wave_reduce_sum(v * v);
        if (lane == 0) atomicAdd(&s_ss, part);
    }
    __syncthreads();
    const float gnn_scale = 1.0f / fmaxf(sqrtf(s_ss), L2_EPS);

    // ---- 3b) waves 0..7: normalize prompt row p, emit output row T+p ----
    if (wid < P_TOK) {
        const int p = wid;
        const float* prow = prompt + p * H_DIM;
        float ss = 0.f;
        for (int j = lane; j < H_DIM; j += 32) { const float x = prow[j]; ss += x * x; }
        ss = wave_reduce_sum(ss);
        const float sc = 1.0f / fmaxf(sqrtf(ss), L2_EPS);

        float* orow = out + (size_t)(T_TOK + p) * OUT_W;
        for (int j = lane; j < H_DIM; j += 32) orow[j] = prow[j] * sc;
        for (int g = lane; g < G_DIM; g += 32) orow[H_DIM + g] = s_gnn[g] * gnn_scale;
    }
}

// ---------------------------------------------------------------------------
// Kernel B: inverse L2 norms of the 4096 sentence rows -> d_ws
// 1 wave per row, 8 waves/block, 512 blocks. Regular-temporal loads so the
// 8 MB sentence table lands (and stays) in the 192 MB L2 for the gather phase.
// ---------------------------------------------------------------------------
__global__ void __launch_bounds__(256)
sent_norm_kernel(const float* __restrict__ sent,   // [N_SENT, G]
                 float* __restrict__ inv_norm)     // [N_SENT]
{
    const int wid  = threadIdx.x >> 5;
    const int lane = threadIdx.x & 31;
    const int row  = blockIdx.x * 8 + wid;
    if (row >= N_SENT) return;

    const v4f* r = (const v4f*)(sent + (size_t)row * G_DIM);
    float ss = 0.f;
    #pragma unroll
    for (int i = 0; i < 4; ++i) {                 // 4 * 32 * 4 = 512 floats
        const v4f v = r[i * 32 + lane];
        ss += v.x * v.x + v.y * v.y + v.z * v.z + v.w * v.w;
    }
    ss = wave_reduce_sum(ss);
    if (lane == 0) inv_norm[row] = 1.0f / fmaxf(sqrtf(ss), L2_EPS);
}

// ---------------------------------------------------------------------------
// Kernel C: main bandwidth kernel. 1 wave per token row.
// out[row] = [ tok_row / ||tok_row|| , sent[id[row]] * inv_norm[id[row]] ]
//
// Cache policy: tok_embs is read-once and out is write-once -> non-temporal
// (TH=NT) so 403 MB of streaming reads + 671 MB of streaming writes don't
// evict the L2-resident 8 MB sentence table that the gather re-reads ~33x.
// ---------------------------------------------------------------------------
__global__ void __launch_bounds__(256)
token_stream_kernel(const float* __restrict__ tok,       // [T, H]
                    const float* __restrict__ sent,      // [N_SENT, G]
                    const int*   __restrict__ ids,       // [T]
                    const float* __restrict__ inv_norm,  // [N_SENT]
                    float* __restrict__ out)             // [T+P, OUT_W]
{
    const int wid  = threadIdx.x >> 5;
    const int lane = threadIdx.x & 31;
    const size_t row = (size_t)blockIdx.x * 8 + wid;
    if (row >= T_TOK) return;

    // --- load the 768-float token row (6x b128 per lane, coalesced, NT) ---
    const v4f* trow = (const v4f*)(tok + row * H_DIM);
    v4f v[6];
    float ss = 0.f;
    #pragma unroll
    for (int i = 0; i < 6; ++i) {
        v[i] = __builtin_nontemporal_load(&trow[i * 32 + lane]);
        ss += v[i].x * v[i].x + v[i].y * v[i].y + v[i].z * v[i].z + v[i].w * v[i].w;
    }
    ss = wave_reduce_sum(ss);
    const float sc = 1.0f / fmaxf(sqrtf(ss), L2_EPS);

    v4f* orow = (v4f*)(out + row * OUT_W);
    #pragma unroll
    for (int i = 0; i < 6; ++i) {
        __builtin_nontemporal_store(v[i] * sc, &orow[i * 32 + lane]);
    }

    // --- gathered sentence row: regular-temporal loads (keep L2-resident) ---
    const int sid = ids[row];
    const float isc = inv_norm[sid];
    const v4f* srow = (const v4f*)(sent + (size_t)sid * G_DIM);
    v4f* orow2 = (v4f*)(out + row * OUT_W + H_DIM);
    #pragma unroll
    for (int i = 0; i < 4; ++i) {
        const v4f g = srow[i * 32 + lane];
        __builtin_nontemporal_store(g * isc, &orow2[i * 32 + lane]);
    }
}

// ---------------------------------------------------------------------------
extern "C" void kernel_launch(void* const* d_in, const int* in_sizes, int n_in,
                              void* d_out, int out_size, void* d_ws, size_t ws_size,
                              hipStream_t stream) {
    const float* tok    = (const float*)d_in[0];  // tok_embs       [T, H]
    const float* sent   = (const float*)d_in[1];  // gnn_sent_embs  [N_SENT, G]
    const float* prompt = (const float*)d_in[2];  // prompt_tokens  [P, H]
    const float* W      = (const float*)d_in[3];  // W              [G, H]
    const float* b      = (const float*)d_in[4];  // b              [G]
    const int*   ids    = (const int*)d_in[5];    // tok_sent_ids   [T]
    float* out = (float*)d_out;
    float* inv_norm = (float*)d_ws;               // N_SENT floats of scratch

    // A: prompt mean + WMMA matvec + prompt output rows (1 WG, 32 waves)
    prompt_wmma_kernel<<<1, 1024, 0, stream>>>(prompt, W, b, out);

    // B: sentence-row inverse norms (4096 rows, 1 wave/row)
    sent_norm_kernel<<<N_SENT / 8, 256, 0, stream>>>(sent, inv_norm);

    // C: main streaming kernel (131072 rows, 1 wave/row)
    token_stream_kernel<<<T_TOK / 8, 256, 0, stream>>>(tok, sent, ids, inv_norm, out);
}